// loss_fn_90709709291733
// MI455X (gfx1250) — compile-verified
//
#include <hip/hip_runtime.h>
#include <hip/hip_bf16.h>

// YOLO loss: bandwidth-bound streaming reduction.
// S=7, B=2, C=2, N=12. rows = 32768*7*7. Output: single f32 scalar.

typedef __attribute__((ext_vector_type(2))) float v2f;
typedef __attribute__((ext_vector_type(8))) float v8f;

#define NCOL 12
#define INV_S (1.0f / 7.0f)
#define LAM_COORD 5.0f
#define LAM_NOOBJ 0.5f
#define NBLOCKS 1024
#define NTHREADS 256

// ---------------------------------------------------------------------------
// Wave32 full reduction. Primary path: one V_WMMA_F32_16X16X4_F32 with
// A = lane partials (16x4, K=1/K=3 zeroed), B = ones. D[m][n] = p[m]+p[m+16].
// Summing the 8 D VGPRs gives half-wave totals; one shfl_xor(16) finishes.
// EXEC must be all ones at the call site (called post-loop, uniformly).
// ---------------------------------------------------------------------------
__device__ __forceinline__ float wave_sum32(float v) {
#if __has_builtin(__builtin_amdgcn_wmma_f32_16x16x4_f32)
  v2f a;
  a.x = v;      // A vgpr0: lanes 0-15 -> (M=lane,K=0); lanes 16-31 -> (M=lane-16,K=2)
  a.y = 0.0f;   // A vgpr1: K=1 / K=3 contribute nothing
  v2f b;
  b.x = 1.0f;   // B = all ones (layout-agnostic)
  b.y = 1.0f;
  v8f c = {};
  v8f d = __builtin_amdgcn_wmma_f32_16x16x4_f32(
      /*neg_a=*/false, a, /*neg_b=*/false, b,
      /*c_mod=*/(short)0, c, /*reuse_a=*/false, /*reuse_b=*/false);
  float s = ((d[0] + d[1]) + (d[2] + d[3])) + ((d[4] + d[5]) + (d[6] + d[7]));
  s += __shfl_xor(s, 16, 32);   // combine the two half-wave totals
  return s;
#else
  for (int off = 16; off > 0; off >>= 1) v += __shfl_xor(v, off, 32);
  return v;
#endif
}

// ---------------------------------------------------------------------------
// Per-cell loss (one grid row of the reference's reshape(-1, 12)).
// ---------------------------------------------------------------------------
__device__ __forceinline__ float cell_loss(const float* __restrict__ p,
                                           const float* __restrict__ l) {
  const float objf = (l[4] > 0.0f) ? 1.0f : 0.0f;
  const float noobjf = 1.0f - objf;

  // label box
  const float lx = l[0], ly = l[1], lw = l[2], lh = l[3];
  const float lcx = lx * INV_S, lcy = ly * INV_S;
  const float l_lo_x = lcx - 0.5f * lw, l_lo_y = lcy - 0.5f * lh;
  const float l_hi_x = lcx + 0.5f * lw, l_hi_y = lcy + 0.5f * lh;
  const float area_l = lw * lh;

  float iou[2], coord_se[2], conf[2];
#pragma unroll
  for (int bb = 0; bb < 2; ++bb) {
    const float* bp = p + 5 * bb;
    const float x = bp[0], y = bp[1], w = bp[2], h = bp[3];
    conf[bb] = bp[4];
    const float cx = x * INV_S, cy = y * INV_S;
    const float p_lo_x = cx - 0.5f * w, p_lo_y = cy - 0.5f * h;
    const float p_hi_x = cx + 0.5f * w, p_hi_y = cy + 0.5f * h;
    float iw = fminf(p_hi_x, l_hi_x) - fmaxf(p_lo_x, l_lo_x);
    float ih = fminf(p_hi_y, l_hi_y) - fmaxf(p_lo_y, l_lo_y);
    iw = fmaxf(iw, 0.0f);
    ih = fmaxf(ih, 0.0f);
    const float inter = iw * ih;
    const float area_p = w * h;           // p_hi - p_lo == (w, h)
    iou[bb] = inter / (area_p + area_l - inter);
    const float dx = x - lx, dy = y - ly, dw = w - lw, dh = h - lh;
    coord_se[bb] = dx * dx + dy * dy + dw * dw + dh * dh;
  }

  // argmax with first-max tie rule (box 0 wins on iou0 >= iou1)
  const int r = (iou[1] > iou[0]) ? 1 : 0;
  const float max_iou = fmaxf(iou[0], iou[1]);

  const float coord = coord_se[r];
  const float dc = conf[r] - max_iou;
  const float conf_resp = dc * dc;
  const float co = conf[1 - r];
  const float conf_not_resp = co * co;

  const float d4 = p[4] - l[4], d9 = p[9] - l[9];
  const float noobj = d4 * d4 + d9 * d9;

  const float d10 = p[10] - l[10], d11 = p[11] - l[11];
  const float cls = d10 * d10 + d11 * d11;

  return objf * (LAM_COORD * coord + conf_resp + LAM_NOOBJ * conf_not_resp + cls)
       + noobjf * (LAM_NOOBJ * noobj);
}

// ---------------------------------------------------------------------------
// Kernel 1: grid-stride over cells, per-block partial to d_ws.
// Rows are 48 bytes and 16B-aligned -> 3x b128 loads per array per row.
// ---------------------------------------------------------------------------
__global__ void yolo_loss_partial(const float* __restrict__ pred,
                                  const float* __restrict__ label,
                                  float* __restrict__ partials, int rows) {
  float acc = 0.0f;
  const int stride = gridDim.x * blockDim.x;
  for (int i = blockIdx.x * blockDim.x + threadIdx.x; i < rows; i += stride) {
    const float4* pp = (const float4*)(pred + (size_t)i * NCOL);
    const float4* lp = (const float4*)(label + (size_t)i * NCOL);

    // speculative prefetch of the next grid-stride iteration (TH=RT spec:
    // OOB translations silently dropped, so no guard needed)
    __builtin_prefetch(pred + ((size_t)i + stride) * NCOL, 0, 0);
    __builtin_prefetch(label + ((size_t)i + stride) * NCOL, 0, 0);

    const float4 p0 = pp[0], p1 = pp[1], p2 = pp[2];
    const float4 l0 = lp[0], l1 = lp[1], l2 = lp[2];
    const float p[NCOL] = {p0.x, p0.y, p0.z, p0.w, p1.x, p1.y,
                           p1.z, p1.w, p2.x, p2.y, p2.z, p2.w};
    const float l[NCOL] = {l0.x, l0.y, l0.z, l0.w, l1.x, l1.y,
                           l1.z, l1.w, l2.x, l2.y, l2.z, l2.w};
    acc += cell_loss(p, l);
  }

  // all lanes reconverged here: EXEC all-1s -> WMMA-legal
  const float wsum = wave_sum32(acc);

  __shared__ float sm[NTHREADS / 32];
  const int wid = threadIdx.x >> 5;
  const int lane = threadIdx.x & 31;
  if (lane == 0) sm[wid] = wsum;
  __syncthreads();
  if (threadIdx.x == 0) {
    float t = 0.0f;
    const int nw = blockDim.x >> 5;
    for (int w = 0; w < nw; ++w) t += sm[w];
    partials[blockIdx.x] = t;
  }
}

// ---------------------------------------------------------------------------
// Kernel 2: single block folds NBLOCKS partials into the scalar output.
// Deterministic (no atomics) so graph replays reproduce bit-identically.
// ---------------------------------------------------------------------------
__global__ void yolo_loss_finalize(const float* __restrict__ partials, int n,
                                   float* __restrict__ out) {
  float acc = 0.0f;
  for (int i = threadIdx.x; i < n; i += blockDim.x) acc += partials[i];

  const float wsum = wave_sum32(acc);

  __shared__ float sm[NTHREADS / 32];
  const int wid = threadIdx.x >> 5;
  const int lane = threadIdx.x & 31;
  if (lane == 0) sm[wid] = wsum;
  __syncthreads();
  if (threadIdx.x == 0) {
    float t = 0.0f;
    const int nw = blockDim.x >> 5;
    for (int w = 0; w < nw; ++w) t += sm[w];
    out[0] = t;
  }
}

extern "C" void kernel_launch(void* const* d_in, const int* in_sizes, int n_in,
                              void* d_out, int out_size, void* d_ws, size_t ws_size,
                              hipStream_t stream) {
  const float* pred = (const float*)d_in[0];
  const float* label = (const float*)d_in[1];
  float* out = (float*)d_out;
  float* partials = (float*)d_ws;  // NBLOCKS * 4 = 4 KB of scratch

  const int rows = in_sizes[0] / NCOL;  // 32768 * 7 * 7

  yolo_loss_partial<<<NBLOCKS, NTHREADS, 0, stream>>>(pred, label, partials, rows);
  yolo_loss_finalize<<<1, NTHREADS, 0, stream>>>(partials, NBLOCKS, out);
}